// _BrunelRNN_1941325217860
// MI455X (gfx1250) — compile-verified
//
#include <hip/hip_runtime.h>

// ---------------------------------------------------------------------------
// Brunel spiking RNN step on MI455X (gfx1250).
//   i_total[b,n] = sum_k spikes[t-15][b,k] * W[n,k] + ext[t,b,n]
//   v' = DECAY*v + i_total ; s = (v' >= 1) ; v = s ? 0 : v'
// GEMM is [M=16(8 batch + 8 zero pad)] x [N=5000] x [K=5000] in fp32 using
// V_WMMA_F32_16X16X4_F32. W (100 MB) lives in the 192 MB L2 -> the kernel is
// L2-bandwidth bound; WMMA keeps the MACs off the VALU. Spike staging is one
// wave per batch row with b128 moves (no integer division).
// ---------------------------------------------------------------------------

typedef __attribute__((ext_vector_type(2))) float v2f;
typedef __attribute__((ext_vector_type(4))) float v4f;
typedef __attribute__((ext_vector_type(8))) float v8f;

#define N_TOT    5000
#define BATCH    8
#define TSTEPS   256
#define DELAY    15
#define DECAY_F  0.95f
#define KCHUNK   1000           // K staged in LDS per chunk (5 chunks)
#define NCHUNK   5
#define WGSZ     256            // 8 wave32
#define WPW      8              // waves per workgroup
#define NTILES   313            // ceil(5000/16)
#define NWG      40             // 40*8 = 320 waves >= 313 tiles

__global__ __launch_bounds__(WGSZ) void brunel_step(
    const float* __restrict__ W,        // [N, N] row-major (W[n, k])
    const float* __restrict__ ext,      // [T, B, N]
    const float* __restrict__ delayed,  // spikes[t-DELAY] base [B, N] (unused if !use_rec)
    float* __restrict__ spk_t,          // out spikes[t] [B, N]
    float* __restrict__ vs_t,           // out vs[t]     [B, N]
    float* __restrict__ vstate,         // persistent v  [B, N] (in d_ws)
    int t, int use_rec, int read_v)
{
    // LDS: 8 rows of spikes [b][kk] + one zeroed row feeding A-pad lanes (m>=8)
    __shared__ float sA[BATCH * KCHUNK + KCHUNK + 4];
    const int tid    = threadIdx.x;
    const int lane   = tid & 31;
    const int waveid = tid >> 5;
    const int m      = lane & 15;            // A: batch row / B: neuron col in tile
    const int koff   = (lane >> 4) << 1;     // 0 for lanes 0-15, 2 for lanes 16-31
    const int tile   = blockIdx.x * WPW + waveid;
    const int n0     = tile * 16;

    // B-operand row pointer (clamped for the partial last tile; results discarded)
    int nrow = n0 + m;
    if (nrow > N_TOT - 1) nrow = N_TOT - 1;
    const float* __restrict__ wrow = W + (size_t)nrow * N_TOT;

    v8f acc0 = {0.f,0.f,0.f,0.f,0.f,0.f,0.f,0.f};
    v8f acc1 = {0.f,0.f,0.f,0.f,0.f,0.f,0.f,0.f};

    const int zbase = BATCH * KCHUNK;
    for (int i = tid; i < KCHUNK + 4; i += WGSZ)
        sA[zbase + i] = 0.f;                 // zero pad row for M=8..15

    if (use_rec) {
        // Per-lane, loop-invariant A base pointer: batches point at their spike row,
        // pad rows point at the zero row -> no per-iteration masking, EXEC
        // stays all-ones across every WMMA (ISA requirement).
        const float* __restrict__ abase =
            (m < BATCH) ? (sA + m * KCHUNK + koff) : (sA + zbase + koff);

        // Wave w stages batch row w: both sides 16B-aligned (5000*4 and
        // kbase*4 are multiples of 16) -> global_load_b128 / ds_store_b128.
        const v4f* __restrict__ srow = (const v4f*)(delayed + (size_t)waveid * N_TOT);
        v4f* __restrict__ drow = (v4f*)(sA + waveid * KCHUNK);

        for (int c = 0; c < NCHUNK; ++c) {
            const int kbase  = c * KCHUNK;
            const int kbase4 = kbase >> 2;   // in float4 units
            __syncthreads();
            for (int q = lane; q < (KCHUNK >> 2); q += 32)   // 250 float4 per row
                drow[q] = srow[kbase4 + q];
            __syncthreads();

            if (tile < NTILES) {
                const float* __restrict__ bbase = wrow + kbase + koff;
                // 125 iters x 2 WMMA: K covered 8 per iter
                // (lanes 0-15 carry K+0,K+1; lanes 16-31 carry K+2,K+3)
                for (int kk = 0; kk < KCHUNK; kk += 8) {
                    v2f a0 = *(const v2f*)(abase + kk);      // ds_load_b64
                    v2f b0 = *(const v2f*)(bbase + kk);      // global_load_b64 (L2)
                    acc0 = __builtin_amdgcn_wmma_f32_16x16x4_f32(
                        false, a0, false, b0, (short)0, acc0, false, false);
                    v2f a1 = *(const v2f*)(abase + kk + 4);
                    v2f b1 = *(const v2f*)(bbase + kk + 4);
                    acc1 = __builtin_amdgcn_wmma_f32_16x16x4_f32(
                        false, a1, false, b1, (short)0, acc1, false, false);
                }
            }
        }
    }

    acc0 = acc0 + acc1;

    // Epilogue: C/D layout -> lanes 0-15 hold M = VGPR index = batch 0..7,
    // N = n0 + lane. Leak, threshold, hard reset; write spikes/vs/state.
    if (tile < NTILES && lane < 16) {
        const int n = n0 + lane;
        if (n < N_TOT) {
            const size_t extb = ((size_t)t * BATCH) * N_TOT + n;
#pragma unroll
            for (int r = 0; r < BATCH; ++r) {
                float i_tot = acc0[r] + ext[extb + (size_t)r * N_TOT];
                float vold  = read_v ? vstate[(size_t)r * N_TOT + n] : 0.f;
                float vnew  = vold * DECAY_F + i_tot;
                float s     = (vnew >= 1.0f) ? 1.0f : 0.0f;
                float vout  = (vnew >= 1.0f) ? 0.0f : vnew;
                spk_t[(size_t)r * N_TOT + n]  = s;
                vs_t[(size_t)r * N_TOT + n]   = vout;
                vstate[(size_t)r * N_TOT + n] = vout;
            }
        }
    }
}

extern "C" void kernel_launch(void* const* d_in, const int* in_sizes, int n_in,
                              void* d_out, int out_size, void* d_ws, size_t ws_size,
                              hipStream_t stream) {
    (void)in_sizes; (void)n_in; (void)out_size; (void)ws_size;
    const float* ext = (const float*)d_in[0];   // [T, B, N]
    const float* W   = (const float*)d_in[1];   // [N, N]
    float* spikes = (float*)d_out;                              // [T, B, N]
    float* vs     = spikes + (size_t)TSTEPS * BATCH * N_TOT;    // [T, B, N]
    float* vstate = (float*)d_ws;                               // [B, N]

    dim3 grid(NWG), block(WGSZ);
    const size_t stepN = (size_t)BATCH * N_TOT;
    for (int t = 0; t < TSTEPS; ++t) {
        const float* delayed = (t >= DELAY) ? (spikes + (size_t)(t - DELAY) * stepN)
                                            : (const float*)nullptr;
        brunel_step<<<grid, block, 0, stream>>>(
            W, ext, delayed,
            spikes + (size_t)t * stepN,
            vs     + (size_t)t * stepN,
            vstate,
            t, (t >= DELAY) ? 1 : 0, (t > 0) ? 1 : 0);
    }
}